// PitchPulseGAT_24945170055514
// MI455X (gfx1250) — compile-verified
//
#include <hip/hip_runtime.h>
#include <math.h>

typedef float v2f __attribute__((ext_vector_type(2)));
typedef float v8f __attribute__((ext_vector_type(8)));

#define NN 50000
#define NE 800000
#define NG 64

__device__ __forceinline__ float leaky(float x) { return x >= 0.0f ? x : 0.2f * x; }
__device__ __forceinline__ float elu1(float x)  { return x > 0.0f ? x : expm1f(x); }

// monotonic float <-> uint mapping so atomicMax(uint) == max(float)
__device__ __forceinline__ unsigned f2o(float f) {
  unsigned u = __float_as_uint(f);
  return (u & 0x80000000u) ? ~u : (u | 0x80000000u);
}
__device__ __forceinline__ float o2f(unsigned o) {
  return __uint_as_float((o & 0x80000000u) ? (o ^ 0x80000000u) : ~o);
}

// ---------------- generic helpers ----------------

__global__ void k_zero(float* p, int n) {
  int t = blockIdx.x * blockDim.x + threadIdx.x;
  if (t < n) p[t] = 0.0f;
}

// hW1[n][j] = x[n][0]*W1[0][j] + x[n][1]*W1[1][j]   (K=2: scalar is optimal)
__global__ void k_gemm1(const float* __restrict__ x, const float* __restrict__ W1,
                        float* __restrict__ hW1) {
  int t = blockIdx.x * blockDim.x + threadIdx.x;
  if (t >= NN * 256) return;
  int n = t >> 8, j = t & 255;
  hW1[t] = x[2 * n] * W1[j] + x[2 * n + 1] * W1[256 + j];
}

// wave-per-node: alpha_src[n][h] = dot(h[n,h,:], a_src[h,:])  (64-wide dot, 32 lanes)
template <int H>
__global__ void k_alpha(const float* __restrict__ h, const float* __restrict__ a_src,
                        const float* __restrict__ a_dst, float* __restrict__ as,
                        float* __restrict__ ad) {
  int wid  = (blockIdx.x * blockDim.x + threadIdx.x) >> 5;
  int lane = threadIdx.x & 31;
  if (wid >= NN) return;
  const int F = H * 64;
  for (int hh = 0; hh < H; ++hh) {
    float v0 = h[(size_t)wid * F + hh * 64 + lane];
    float v1 = h[(size_t)wid * F + hh * 64 + 32 + lane];
    float ps = v0 * a_src[hh * 64 + lane] + v1 * a_src[hh * 64 + 32 + lane];
    float pd = v0 * a_dst[hh * 64 + lane] + v1 * a_dst[hh * 64 + 32 + lane];
    for (int off = 16; off; off >>= 1) {
      ps += __shfl_xor(ps, off, 32);
      pd += __shfl_xor(pd, off, 32);
    }
    if (lane == 0) { as[wid * H + hh] = ps; ad[wid * H + hh] = pd; }
  }
}

// m[n][h] = ord(self-loop e)  (every node has a self loop -> valid max seed)
template <int H>
__global__ void k_init_m(const float* __restrict__ as, const float* __restrict__ ad,
                         unsigned* __restrict__ m) {
  int t = blockIdx.x * blockDim.x + threadIdx.x;
  if (t >= NN * H) return;
  m[t] = f2o(leaky(as[t] + ad[t]));
}

// per (edge, head): e = leaky(as[src]+ad[dst]); store; atomicMax into m[dst]
template <int H>
__global__ void k_edge_max(const int* __restrict__ ei, const float* __restrict__ as,
                           const float* __restrict__ ad, float* __restrict__ ev,
                           unsigned* __restrict__ m) {
  int t = blockIdx.x * blockDim.x + threadIdx.x;
  if (t >= NE * H) return;
  int e = t / H, hh = t % H;
  int s = ei[e], d = ei[NE + e];
  float v = leaky(as[s * H + hh] + ad[d * H + hh]);
  ev[t] = v;
  atomicMax(&m[d * H + hh], f2o(v));
}

// s[n][h] = exp(e_self - m)   (self-loop numerator seeds the sum)
template <int H>
__global__ void k_init_s(const float* __restrict__ as, const float* __restrict__ ad,
                         const unsigned* __restrict__ m, float* __restrict__ s) {
  int t = blockIdx.x * blockDim.x + threadIdx.x;
  if (t >= NN * H) return;
  s[t] = expf(leaky(as[t] + ad[t]) - o2f(m[t]));
}

// per (edge, head): ex = exp(e - m[dst]); ev := ex; atomicAdd s[dst]
template <int H>
__global__ void k_edge_sum(const int* __restrict__ ei, float* __restrict__ ev,
                           const unsigned* __restrict__ m, float* __restrict__ s) {
  int t = blockIdx.x * blockDim.x + threadIdx.x;
  if (t >= NE * H) return;
  int e = t / H, hh = t % H;
  int d = ei[NE + e];
  float ex = expf(ev[t] - o2f(m[d * H + hh]));
  ev[t] = ex;
  atomicAdd(&s[d * H + hh], ex);
}

// out[n][h*64+c] = h[n][h*64+c] * alpha_self   (initializes the segment sum)
template <int H>
__global__ void k_init_out(const float* __restrict__ hmat, const float* __restrict__ as,
                           const float* __restrict__ ad, const unsigned* __restrict__ m,
                           const float* __restrict__ s, float* __restrict__ out) {
  const int F = H * 64;
  int t = blockIdx.x * blockDim.x + threadIdx.x;
  if (t >= NN * F) return;
  int n = t / F, j = t % F, hh = j >> 6;
  float es  = leaky(as[n * H + hh] + ad[n * H + hh]);
  float num = expf(es - o2f(m[n * H + hh]));
  float al  = num / (s[n * H + hh] + 1e-16f);
  out[t] = hmat[t] * al;
}

// per (edge, channel): scatter h[src]*alpha into out[dst] for each head
template <int H>
__global__ void k_message(const int* __restrict__ ei, const float* __restrict__ hmat,
                          const float* __restrict__ ev, const float* __restrict__ s,
                          float* __restrict__ out) {
  const int F = H * 64;
  long t = (long)blockIdx.x * blockDim.x + threadIdx.x;
  if (t >= (long)NE * 64) return;
  int e = (int)(t >> 6), c = (int)(t & 63);
  int sn = ei[e], d = ei[NE + e];
  for (int hh = 0; hh < H; ++hh) {
    float w = ev[e * H + hh] / (s[d * H + hh] + 1e-16f);
    atomicAdd(&out[(size_t)d * F + hh * 64 + c],
              hmat[(size_t)sn * F + hh * 64 + c] * w);
  }
}

// out := elu(out + bias)  in place
__global__ void k_bias_elu(float* __restrict__ out, const float* __restrict__ b, int F,
                           long total) {
  long t = (long)blockIdx.x * blockDim.x + threadIdx.x;
  if (t >= total) return;
  out[t] = elu1(out[t] + b[(int)(t % F)]);
}

// ---------------- WMMA f32 GEMM: hW2 = h1[50000,256] @ W2[256,64] ----------------

// swizzle W2 into per-fragment layout: W2s[ks][nt][lane][v] = W2[ks*4 + v + 2*(lane/16)][nt*16 + lane%16]
__global__ void k_w2swz(const float* __restrict__ W2, float* __restrict__ W2s) {
  int t = blockIdx.x * blockDim.x + threadIdx.x;
  if (t >= 64 * 4 * 32 * 2) return;
  int v = t & 1, lane = (t >> 1) & 31, nt = (t >> 6) & 3, ks = t >> 8;
  int g = lane >> 4, ln = lane & 15;
  int K = ks * 4 + v + 2 * g;
  W2s[t] = W2[K * 64 + nt * 16 + ln];
}

// 256 threads = 8 waves; each wave computes a 16x64 tile via V_WMMA_F32_16X16X4_F32
__global__ void __launch_bounds__(256) k_gemm2(const float* __restrict__ h1,
                                               const float* __restrict__ W2s,
                                               float* __restrict__ hW2) {
  int w = threadIdx.x >> 5, lane = threadIdx.x & 31;
  int g = lane >> 4, ln = lane & 15;
  int rowBase = blockIdx.x * 128 + w * 16;
  int ar = rowBase + ln;
  if (ar > NN - 1) ar = NN - 1;                 // clamp reads; stores guarded below
  const float* arow = h1 + (size_t)ar * 256;

  v8f acc[4] = {v8f{}, v8f{}, v8f{}, v8f{}};
  for (int ks = 0; ks < 64; ++ks) {
    // A fragment: lane%16 = M row; VGPR v holds K = ks*4 + v + 2*(lane/16) (ISA 16x4 f32 layout)
    v2f a = *(const v2f*)(arow + ks * 4 + 2 * g);
    const float* bp = W2s + ks * 256 + lane * 2;
    v2f b0 = *(const v2f*)(bp);
    v2f b1 = *(const v2f*)(bp + 64);
    v2f b2 = *(const v2f*)(bp + 128);
    v2f b3 = *(const v2f*)(bp + 192);
    acc[0] = __builtin_amdgcn_wmma_f32_16x16x4_f32(false, a, false, b0, (short)0, acc[0], false, false);
    acc[1] = __builtin_amdgcn_wmma_f32_16x16x4_f32(false, a, false, b1, (short)0, acc[1], false, false);
    acc[2] = __builtin_amdgcn_wmma_f32_16x16x4_f32(false, a, false, b2, (short)0, acc[2], false, false);
    acc[3] = __builtin_amdgcn_wmma_f32_16x16x4_f32(false, a, false, b3, (short)0, acc[3], false, false);
  }
  // C/D layout: lane%16 = N, VGPR r -> M = r + 8*(lane/16)
  for (int nt = 0; nt < 4; ++nt)
    for (int r = 0; r < 8; ++r) {
      int row = rowBase + r + 8 * g;
      if (row < NN) hW2[(size_t)row * 64 + nt * 16 + ln] = acc[nt][r];
    }
}

// ---------------- pooling + head ----------------

__global__ void k_pool(const float* __restrict__ h2, const int* __restrict__ batch,
                       float* __restrict__ pooled, float* __restrict__ cnt) {
  long t = (long)blockIdx.x * blockDim.x + threadIdx.x;
  if (t >= (long)NN * 64) return;
  int n = (int)(t >> 6), c = (int)(t & 63);
  int b = batch[n];
  atomicAdd(&pooled[b * 64 + c], h2[t]);
  if (c == 0) atomicAdd(&cnt[b], 1.0f);
}

__global__ void k_final(const float* __restrict__ pooled, const float* __restrict__ cnt,
                        const float* __restrict__ W3, const float* __restrict__ b3,
                        float* __restrict__ out) {
  int gid = threadIdx.x;
  if (gid >= NG) return;
  float c = fmaxf(cnt[gid], 1.0f);
  float acc = 0.0f;
  for (int j = 0; j < 64; ++j) acc += (pooled[gid * 64 + j] / c) * W3[j];
  out[gid] = acc + b3[0];
}

// ---------------- launcher ----------------

static inline int gdiv(long n, int b) { return (int)((n + b - 1) / b); }

extern "C" void kernel_launch(void* const* d_in, const int* in_sizes, int n_in,
                              void* d_out, int out_size, void* d_ws, size_t ws_size,
                              hipStream_t stream) {
  const float* x     = (const float*)d_in[0];
  const int*   ei    = (const int*)d_in[1];    // [2, NE]: src row then dst row
  const int*   batch = (const int*)d_in[2];
  const float* W1    = (const float*)d_in[3];
  const float* asrc1 = (const float*)d_in[4];
  const float* adst1 = (const float*)d_in[5];
  const float* b1    = (const float*)d_in[6];
  const float* W2    = (const float*)d_in[7];
  const float* asrc2 = (const float*)d_in[8];
  const float* adst2 = (const float*)d_in[9];
  const float* b2    = (const float*)d_in[10];
  const float* W3    = (const float*)d_in[11];
  const float* b3    = (const float*)d_in[12];
  float* out = (float*)d_out;

  float* ws = (float*)d_ws;
  size_t off = 0;
  auto alloc = [&](size_t n) { float* p = ws + off; off += n; return p; };
  float*    hW1  = alloc((size_t)NN * 256);
  float*    out1 = alloc((size_t)NN * 256);
  float*    as1  = alloc((size_t)NN * 4);
  float*    ad1  = alloc((size_t)NN * 4);
  unsigned* m1   = (unsigned*)alloc((size_t)NN * 4);
  float*    s1   = alloc((size_t)NN * 4);
  float*    e1   = alloc((size_t)NE * 4);
  float*    hW2  = alloc((size_t)NN * 64);
  float*    out2 = alloc((size_t)NN * 64);
  float*    W2s  = alloc(64 * 4 * 32 * 2);
  float*    as2  = alloc(NN);
  float*    ad2  = alloc(NN);
  unsigned* m2   = (unsigned*)alloc(NN);
  float*    s2   = alloc(NN);
  float*    e2   = alloc(NE);
  float*    pooled = alloc(NG * 64);
  float*    cnt    = alloc(NG);

  const int B = 256;

  k_zero<<<gdiv(NG * 64 + NG, B), B, 0, stream>>>(pooled, NG * 64 + NG);

  // ---- layer 1 (H=4, F=256) ----
  k_gemm1<<<gdiv((long)NN * 256, B), B, 0, stream>>>(x, W1, hW1);
  k_alpha<4><<<gdiv((long)NN * 32, B), B, 0, stream>>>(hW1, asrc1, adst1, as1, ad1);
  k_init_m<4><<<gdiv((long)NN * 4, B), B, 0, stream>>>(as1, ad1, m1);
  k_edge_max<4><<<gdiv((long)NE * 4, B), B, 0, stream>>>(ei, as1, ad1, e1, m1);
  k_init_s<4><<<gdiv((long)NN * 4, B), B, 0, stream>>>(as1, ad1, m1, s1);
  k_edge_sum<4><<<gdiv((long)NE * 4, B), B, 0, stream>>>(ei, e1, m1, s1);
  k_init_out<4><<<gdiv((long)NN * 256, B), B, 0, stream>>>(hW1, as1, ad1, m1, s1, out1);
  k_message<4><<<gdiv((long)NE * 64, B), B, 0, stream>>>(ei, hW1, e1, s1, out1);
  k_bias_elu<<<gdiv((long)NN * 256, B), B, 0, stream>>>(out1, b1, 256, (long)NN * 256);

  // ---- layer 2 GEMM: hW2 = h1 @ W2 via V_WMMA_F32_16X16X4_F32 ----
  k_w2swz<<<gdiv(64 * 4 * 32 * 2, B), B, 0, stream>>>(W2, W2s);
  k_gemm2<<<gdiv(NN, 128), B, 0, stream>>>(out1, W2s, hW2);

  // ---- layer 2 attention (H=1, F=64) ----
  k_alpha<1><<<gdiv((long)NN * 32, B), B, 0, stream>>>(hW2, asrc2, adst2, as2, ad2);
  k_init_m<1><<<gdiv(NN, B), B, 0, stream>>>(as2, ad2, m2);
  k_edge_max<1><<<gdiv(NE, B), B, 0, stream>>>(ei, as2, ad2, e2, m2);
  k_init_s<1><<<gdiv(NN, B), B, 0, stream>>>(as2, ad2, m2, s2);
  k_edge_sum<1><<<gdiv(NE, B), B, 0, stream>>>(ei, e2, m2, s2);
  k_init_out<1><<<gdiv((long)NN * 64, B), B, 0, stream>>>(hW2, as2, ad2, m2, s2, out2);
  k_message<1><<<gdiv((long)NE * 64, B), B, 0, stream>>>(ei, hW2, e2, s2, out2);
  k_bias_elu<<<gdiv((long)NN * 64, B), B, 0, stream>>>(out2, b2, 64, (long)NN * 64);

  // ---- pool + head ----
  k_pool<<<gdiv((long)NN * 64, B), B, 0, stream>>>(out2, batch, pooled, cnt);
  k_final<<<1, 64, 0, stream>>>(pooled, cnt, W3, b3, out);
}